// MultiQuerySelfAttentionIncremental_25486335935235
// MI455X (gfx1250) — compile-verified
//
#include <hip/hip_runtime.h>
#include <hip/hip_bf16.h>

// ---------------------------------------------------------------------------
// MQA incremental attention, MI455X (gfx1250, wave32, WMMA).
// Memory-bound (~268 MB @ 23.3 TB/s ~ 11.5us floor): the KV-cache copy is
// fused into the compute kernels so prev_K/prev_V are each read exactly once.
// All GEMMs use V_WMMA_F32_16X16X4_F32 (full f32 precision).
// ---------------------------------------------------------------------------

typedef __attribute__((ext_vector_type(2))) float v2f;
typedef __attribute__((ext_vector_type(8))) float v8f;

#define NB   32      // batch
#define NH   16      // heads
#define DM   2048    // d_model
#define DK   128     // k dim (== v dim)
#define MP   4095    // prev positions
#define MT   4096    // total positions

__device__ __forceinline__ v8f wmma4(v2f a, v2f b, v8f c) {
  // D = A(16x4 f32) * B(4x16 f32) + C(16x16 f32)
  return __builtin_amdgcn_wmma_f32_16x16x4_f32(
      /*neg_a=*/false, a, /*neg_b=*/false, b,
      /*c_mod=*/(short)0, c, /*reuse_a=*/false, /*reuse_b=*/false);
}

// ---------------------------------------------------------------------------
// Kernel A: projections. q = x @ P_q[h]  (16 GEMMs of [32x2048]x[2048x128]),
// new_k = x @ P_k, new_v = x @ P_v written directly into row 4095 of
// new_K / new_V in d_out. One 16x16 tile per wave-sized block.
// ---------------------------------------------------------------------------
__global__ void mqa_proj_kernel(const float* __restrict__ x,
                                const float* __restrict__ Pq,
                                const float* __restrict__ Pk,
                                const float* __restrict__ Pv,
                                float* __restrict__ q,     // ws [32, 2048]
                                float* __restrict__ outK,  // [32, 4096, 128]
                                float* __restrict__ outV) {
  const int lane = threadIdx.x;
  const int blk  = blockIdx.x;   // 0..287
  const float* Bm;
  float* Dst;
  long   ldd;
  int    m0, n0;
  if (blk < 256) {                       // q projection, per head
    int head = blk >> 4, t = blk & 15;
    m0 = (t >> 3) * 16; n0 = (t & 7) * 16;
    Bm  = Pq + (size_t)head * DM * DK;
    Dst = q + head * DK; ldd = DM;       // q[b, h*128 + k]
  } else if (blk < 272) {                // new_k -> new_K row 4095
    int t = blk - 256;
    m0 = (t >> 3) * 16; n0 = (t & 7) * 16;
    Bm  = Pk;
    Dst = outK + (size_t)MP * DK; ldd = (long)MT * DK;
  } else {                               // new_v -> new_V row 4095
    int t = blk - 272;
    m0 = (t >> 3) * 16; n0 = (t & 7) * 16;
    Bm  = Pv;
    Dst = outV + (size_t)MP * DK; ldd = (long)MT * DK;
  }
  const int mm   = lane & 15;
  const int koff = (lane >> 4) * 2;
  const float* ap = x + (size_t)(m0 + mm) * DM + koff;     // A[m][k]
  const float* bp = Bm + (size_t)koff * DK + n0 + mm;      // B[k][n]
  v8f acc = {};
  for (int k = 0; k < DM; k += 4) {
    v2f a, b;
    a.x = ap[k];                 a.y = ap[k + 1];
    b.x = bp[(size_t)k * DK];    b.y = bp[(size_t)(k + 1) * DK];
    acc = wmma4(a, b, acc);
  }
  const int mb = m0 + 8 * (lane >> 4);
#pragma unroll
  for (int r = 0; r < 8; ++r)
    Dst[(size_t)(mb + r) * ldd + n0 + mm] = acc[r];
}

// ---------------------------------------------------------------------------
// Kernel B: per (batch, 16-position tile):
//   * float4 copy prev_K/prev_V -> new_K/new_V (the only read of prev_*),
//     staging the K tile into LDS (row stride 132 to avoid bank conflicts)
//   * logits[b, 0:16 heads, m0:m0+16] = q[b] (16x128) @ Ktile^T via WMMA
// ---------------------------------------------------------------------------
#define TKS 132   // padded LDS row stride (floats)

__global__ void mqa_logits_kernel(const float* __restrict__ q,      // [32,2048]
                                  const float* __restrict__ prevK,  // [32,4095,128]
                                  const float* __restrict__ prevV,
                                  float* __restrict__ outK,         // [32,4096,128]
                                  float* __restrict__ outV,
                                  float* __restrict__ logits) {     // [32,16,4096]
  __shared__ float tk[16 * TKS];
  const int lane = threadIdx.x;
  const int b    = blockIdx.x >> 8;
  const int m0   = (blockIdx.x & 255) * 16;

  // fused copy + LDS staging: 512 float4s per 16x128 tile, 16 per lane
#pragma unroll
  for (int j = 0; j < 16; ++j) {
    int idx = lane + 32 * j;
    int row = idx >> 5, c4 = idx & 31;
    int m   = m0 + row;
    float4 kv;
    if (m < MP) {
      size_t src = ((size_t)b * MP + m) * DK + c4 * 4;
      size_t dst = ((size_t)b * MT + m) * DK + c4 * 4;
      kv = *(const float4*)(prevK + src);
      float4 vv = *(const float4*)(prevV + src);
      *(float4*)(outK + dst) = kv;
      *(float4*)(outV + dst) = vv;
    } else {  // row 4095 already produced by mqa_proj_kernel
      kv = *(const float4*)(outK + ((size_t)b * MT + MP) * DK + c4 * 4);
    }
    *(float4*)(tk + row * TKS + c4 * 4) = kv;
  }
  __syncthreads();

  const int nn   = lane & 15;            // position within tile (B N-dim)
  const int koff = (lane >> 4) * 2;
  const float* qp = q + (size_t)b * DM + nn * DK + koff;   // A[h][k], h=nn role of M
  const float* bp = tk + nn * TKS + koff;                  // B[k][n] = K[n][k]
  v8f acc = {};
#pragma unroll 8
  for (int k = 0; k < DK; k += 4) {
    v2f a, bb;
    a.x  = qp[k]; a.y  = qp[k + 1];
    bb.x = bp[k]; bb.y = bp[k + 1];
    acc = wmma4(a, bb, acc);
  }
  const int hb = 8 * (lane >> 4);
#pragma unroll
  for (int r = 0; r < 8; ++r)
    logits[((size_t)b * NH + hb + r) * (size_t)MT + m0 + nn] = acc[r];
}

// ---------------------------------------------------------------------------
// Kernel C: softmax over M=4096, one block per (b,h) row, in place.
// ---------------------------------------------------------------------------
__global__ void mqa_softmax_kernel(float* __restrict__ logits) {
  float* p = logits + (size_t)blockIdx.x * MT;
  __shared__ float sm[256];
  const int t = threadIdx.x;
  float mx = -3.402823466e38f;
  for (int i = t; i < MT; i += 256) mx = fmaxf(mx, p[i]);
  sm[t] = mx; __syncthreads();
  for (int s = 128; s > 0; s >>= 1) {
    if (t < s) sm[t] = fmaxf(sm[t], sm[t + s]);
    __syncthreads();
  }
  mx = sm[0]; __syncthreads();
  float sum = 0.0f;
  for (int i = t; i < MT; i += 256) {
    float e = __expf(p[i] - mx);
    p[i] = e; sum += e;
  }
  sm[t] = sum; __syncthreads();
  for (int s = 128; s > 0; s >>= 1) {
    if (t < s) sm[t] += sm[t + s];
    __syncthreads();
  }
  float inv = 1.0f / sm[0];
  for (int i = t; i < MT; i += 256) p[i] *= inv;
}

// ---------------------------------------------------------------------------
// Kernel D: O[b] = W[b] (16x4096) @ new_V[b] (4096x128). Reads the completed
// new_V from d_out (no tail-row branch). One 16x16 v-tile per wave.
// ---------------------------------------------------------------------------
__global__ void mqa_attnv_kernel(const float* __restrict__ w,     // [32,16,4096]
                                 const float* __restrict__ newV,  // [32,4096,128]
                                 float* __restrict__ O) {         // [32, 16*128]
  const int lane = threadIdx.x;
  const int b    = blockIdx.x >> 3;
  const int v0   = (blockIdx.x & 7) * 16;
  const int nn   = lane & 15;
  const int koff = (lane >> 4) * 2;
  const float* ap = w + ((size_t)b * NH + nn) * (size_t)MT + koff;       // A[h][m]
  const float* bp = newV + (size_t)b * MT * DK + (size_t)koff * DK + v0 + nn;
  v8f acc = {};
#pragma unroll 2
  for (int k = 0; k < MT; k += 4) {
    v2f a, bb;
    a.x  = ap[k];               a.y  = ap[k + 1];
    bb.x = bp[(size_t)k * DK];  bb.y = bp[(size_t)(k + 1) * DK];
    acc = wmma4(a, bb, acc);
  }
  const int hb = 8 * (lane >> 4);
#pragma unroll
  for (int r = 0; r < 8; ++r)
    O[(size_t)b * (NH * DK) + (hb + r) * DK + v0 + nn] = acc[r];
}

// ---------------------------------------------------------------------------
// Kernel E: Y[b,d] = sum_{h,v} O[b,h,v] * P_o[h,d,v]; 32x2048 GEMM, K=2048
// with c = h*128+v (c,c+1 always share h since steps are even).
// ---------------------------------------------------------------------------
__global__ void mqa_outproj_kernel(const float* __restrict__ O,   // [32,2048]
                                   const float* __restrict__ Po,  // [16,2048,128]
                                   float* __restrict__ Y) {       // [32,2048]
  const int lane = threadIdx.x;
  const int m0   = (blockIdx.x >> 7) * 16;    // batch tile
  const int n0   = (blockIdx.x & 127) * 16;   // d tile
  const int nn   = lane & 15;
  const int koff = (lane >> 4) * 2;
  const float* ap = O + (size_t)(m0 + nn) * (NH * DK) + koff;
  v8f acc = {};
  for (int k = 0; k < NH * DK; k += 4) {
    int c = k + koff;
    int h = c >> 7, v = c & 127;
    const float* bp = Po + ((size_t)h * DM + n0 + nn) * DK + v;
    v2f a, bb;
    a.x  = ap[k]; a.y  = ap[k + 1];
    bb.x = bp[0]; bb.y = bp[1];
    acc = wmma4(a, bb, acc);
  }
  const int mb = m0 + 8 * (lane >> 4);
#pragma unroll
  for (int r = 0; r < 8; ++r)
    Y[(size_t)(mb + r) * DM + n0 + nn] = acc[r];
}

// ---------------------------------------------------------------------------
extern "C" void kernel_launch(void* const* d_in, const int* in_sizes, int n_in,
                              void* d_out, int out_size, void* d_ws, size_t ws_size,
                              hipStream_t stream) {
  const float* x  = (const float*)d_in[0];
  const float* pK = (const float*)d_in[1];
  const float* pV = (const float*)d_in[2];
  const float* Pq = (const float*)d_in[3];
  const float* Pk = (const float*)d_in[4];
  const float* Pv = (const float*)d_in[5];
  const float* Po = (const float*)d_in[6];

  float* outY = (float*)d_out;                       // [32,2048]
  float* outK = outY + (size_t)NB * DM;              // [32,4096,128]
  float* outV = outK + (size_t)NB * MT * DK;         // [32,4096,128]

  float* q  = (float*)d_ws;                          // 65536 floats
  float* lg = q + (size_t)NB * DM;                   // 2097152 floats
  float* O  = lg + (size_t)NB * NH * MT;             // 65536 floats

  mqa_proj_kernel<<<288, 32, 0, stream>>>(x, Pq, Pk, Pv, q, outK, outV);
  mqa_logits_kernel<<<NB * 256, 32, 0, stream>>>(q, pK, pV, outK, outV, lg);
  mqa_softmax_kernel<<<NB * NH, 256, 0, stream>>>(lg);
  mqa_attnv_kernel<<<NB * 8, 32, 0, stream>>>(lg, outV, O);
  mqa_outproj_kernel<<<256, 32, 0, stream>>>(O, Po, outY);
}